// bingham_KL_divergence_50946902065557
// MI455X (gfx1250) — compile-verified
//
#include <hip/hip_runtime.h>
#include <stdint.h>

#define TILE     512
#define NTHREADS 256

typedef __attribute__((ext_vector_type(4))) unsigned int u32x4;
typedef __attribute__((ext_vector_type(8))) int          i32x8;
typedef __attribute__((ext_vector_type(4))) int          i32x4;

#if __has_builtin(__builtin_amdgcn_tensor_load_to_lds)
#define HAS_TDM 1
#else
#define HAS_TDM 0
#endif

__device__ __forceinline__ unsigned lds_off(const void* p) {
  // __shared__ pointers on amdgcn are LDS byte offsets (addrspace(3))
  return (unsigned)(uintptr_t)p;
}

#if HAS_TDM
// Issue one TDM DMA: copy `ndw` dwords from global `g` into LDS offset `ldsOff`.
// D# layout per CDNA5 ISA ch.8 (CDNA5 ISA): 1-row 2-D tile of dwords.
__device__ __forceinline__ void tdm_load_1d(unsigned ldsOff, const float* g, unsigned ndw) {
  u32x4 g0;
  i32x8 g1;
  i32x4 g2;
  i32x4 g3;
  i32x8 g4;  // extra operand group (clang-23 6-arg form); zero-filled
  unsigned long long ga = (unsigned long long)(uintptr_t)g;

  // ---- group 0 (128b): count=1 | lds_addr | global_addr[56:0] | type=2 ----
  g0[0] = 1u;                                              // count=1, user mode
  g0[1] = ldsOff;                                          // lds_addr (bytes)
  g0[2] = (unsigned)(ga & 0xFFFFFFFFu);                    // global_addr[31:0]
  g0[3] = (unsigned)((ga >> 32) & 0x01FFFFFFu)             // global_addr[56:32]
        | 0x80000000u;                                     // type = 2 ("image")

  // ---- group 1 (256b) ----
  g1[0] = 0x00020000;                       // wg_mask=0, data_size=2 (4B), no flags
  g1[1] = (int)((ndw & 0xFFFFu) << 16);     // barrier_addr=0 | tensor_dim0[15:0]
  g1[2] = (int)(((ndw >> 16) & 0xFFFFu)     // tensor_dim0[31:16]
        | (1u << 16));                      // tensor_dim1 = 1
  g1[3] = (int)(ndw << 16);                 // tensor_dim1 hi=0 | tile_dim0 = ndw (<=2048)
  g1[4] = 1;                                // tile_dim1 = 1, tile_dim2 = 0
  g1[5] = (int)ndw;                         // tensor_dim0_stride[31:0]
  g1[6] = 0;                                // stride0 hi | stride1 lo
  g1[7] = 0;                                // stride1 hi

  // ---- group 2/3: benign dims for unused higher dimensions ----
  g2[0] = 1;                                // tensor_dim2 = 1
  g2[1] = 1;                                // tensor_dim3 = 1 (iterate off)
  g2[2] = 0;                                // tensor_dim2_stride lo
  g2[3] = 0;                                // stride hi | tile_dim3 = 0
  g3[0] = 0;                                // tensor_dim3_stride lo
  g3[1] = (int)(1u << 16);                  // stride hi | tensor_dim4 = 1 (lo16)
  g3[2] = 0;                                // tensor_dim4 hi | tile_dim4 = 0
  g3[3] = 0;

  g4[0] = 0; g4[1] = 0; g4[2] = 0; g4[3] = 0;
  g4[4] = 0; g4[5] = 0; g4[6] = 0; g4[7] = 0;

  __builtin_amdgcn_tensor_load_to_lds(g0, g1, g2, g3, g4, 0);
}
#endif

__global__ __launch_bounds__(NTHREADS)
void bingham_kl_kernel(const float* __restrict__ q1g, const float* __restrict__ z1g,
                       const float* __restrict__ q2g, const float* __restrict__ z2g,
                       const float* __restrict__ F1g, const float* __restrict__ dF1g,
                       const float* __restrict__ F2g, float* __restrict__ out, int B) {
  __shared__ float s_q1 [TILE * 4];
  __shared__ float s_q2 [TILE * 4];
  __shared__ float s_z1 [TILE * 3];
  __shared__ float s_z2 [TILE * 3];
  __shared__ float s_dF1[TILE * 3];
  __shared__ float s_F1 [TILE];
  __shared__ float s_F2 [TILE];

  const int base  = (int)blockIdx.x * TILE;
  int nelem = B - base;
  if (nelem > TILE) nelem = TILE;

#if HAS_TDM
  // One wave issues the DMA for the whole workgroup (TDM ignores EXEC, so
  // gate by branching the entire block away for waves 1..7).
  if (threadIdx.x < 32) {
    const unsigned n = (unsigned)nelem;
    tdm_load_1d(lds_off(s_q1),  q1g  + (size_t)base * 4, n * 4u);
    tdm_load_1d(lds_off(s_q2),  q2g  + (size_t)base * 4, n * 4u);
    tdm_load_1d(lds_off(s_z1),  z1g  + (size_t)base * 3, n * 3u);
    tdm_load_1d(lds_off(s_z2),  z2g  + (size_t)base * 3, n * 3u);
    tdm_load_1d(lds_off(s_dF1), dF1g + (size_t)base * 3, n * 3u);
    tdm_load_1d(lds_off(s_F1),  F1g  + (size_t)base,     n);
    tdm_load_1d(lds_off(s_F2),  F2g  + (size_t)base,     n);
#if __has_builtin(__builtin_amdgcn_s_wait_tensorcnt)
    __builtin_amdgcn_s_wait_tensorcnt(0);
#else
    asm volatile("s_wait_tensorcnt 0x0" ::: "memory");
#endif
  }
#else
  // Fallback: cooperative vectorized staging.
  for (int i = threadIdx.x; i < nelem * 4; i += NTHREADS) {
    s_q1[i] = q1g[(size_t)base * 4 + i];
    s_q2[i] = q2g[(size_t)base * 4 + i];
  }
  for (int i = threadIdx.x; i < nelem * 3; i += NTHREADS) {
    s_z1[i]  = z1g [(size_t)base * 3 + i];
    s_z2[i]  = z2g [(size_t)base * 3 + i];
    s_dF1[i] = dF1g[(size_t)base * 3 + i];
  }
  for (int i = threadIdx.x; i < nelem; i += NTHREADS) {
    s_F1[i] = F1g[(size_t)base + i];
    s_F2[i] = F2g[(size_t)base + i];
  }
#endif
  __syncthreads();

  for (int s = threadIdx.x; s < nelem; s += NTHREADS) {
    const float a1 = s_q1[s * 4 + 0], b1 = s_q1[s * 4 + 1];
    const float c1 = s_q1[s * 4 + 2], d1 = s_q1[s * 4 + 3];
    const float a2 = s_q2[s * 4 + 0], b2 = s_q2[s * 4 + 1];
    const float c2 = s_q2[s * 4 + 2], d2 = s_q2[s * 4 + 3];

    // V1_ft is orthogonal for unit q1 => inverse == transpose.
    // Columns of V1_ft: u0=q1, u1=r0(q1), u2=r1(q1), u3=r2(q1).
    const float u0x =  a1, u0y =  b1, u0z =  c1, u0w =  d1;
    const float u1x =  d1, u1y =  c1, u1z = -b1, u1w = -a1;
    const float u2x = -c1, u2y =  d1, u2z =  a1, u2w = -b1;
    const float u3x = -b1, u3y =  a1, u3z = -d1, u3w =  c1;

    // Rows of V2 = quat_basis(q2).
    const float v0x =  d2, v0y =  c2, v0z = -b2, v0w = -a2;
    const float v1x = -c2, v1y =  d2, v1z =  a2, v1w = -b2;
    const float v2x = -b2, v2y =  a2, v2z = -d2, v2w =  c2;

    const float F1v   = s_F1[s];
    const float invF1 = 1.0f / F1v;
    const float dF0 = s_dF1[s * 3 + 0];
    const float dF1_ = s_dF1[s * 3 + 1];
    const float dF2 = s_dF1[s * 3 + 2];
    const float r0 = dF0 * invF1, r1 = dF1_ * invF1, r2 = dF2 * invF1;
    const float w0 = 1.0f - (r0 + r1 + r2);

#define DOT4(ax, ay, az, aw, bx, by, bz, bw) \
    fmaf((ax), (bx), fmaf((ay), (by), fmaf((az), (bz), (aw) * (bw))))

    float H[3];
    {
      const float g0 = DOT4(v0x, v0y, v0z, v0w, u0x, u0y, u0z, u0w);
      const float g1 = DOT4(v0x, v0y, v0z, v0w, u1x, u1y, u1z, u1w);
      const float g2 = DOT4(v0x, v0y, v0z, v0w, u2x, u2y, u2z, u2w);
      const float g3 = DOT4(v0x, v0y, v0z, v0w, u3x, u3y, u3z, u3w);
      H[0] = fmaf(g0 * g0, w0, fmaf(g1 * g1, r0, fmaf(g2 * g2, r1, g3 * g3 * r2)));
    }
    {
      const float g0 = DOT4(v1x, v1y, v1z, v1w, u0x, u0y, u0z, u0w);
      const float g1 = DOT4(v1x, v1y, v1z, v1w, u1x, u1y, u1z, u1w);
      const float g2 = DOT4(v1x, v1y, v1z, v1w, u2x, u2y, u2z, u2w);
      const float g3 = DOT4(v1x, v1y, v1z, v1w, u3x, u3y, u3z, u3w);
      H[1] = fmaf(g0 * g0, w0, fmaf(g1 * g1, r0, fmaf(g2 * g2, r1, g3 * g3 * r2)));
    }
    {
      const float g0 = DOT4(v2x, v2y, v2z, v2w, u0x, u0y, u0z, u0w);
      const float g1 = DOT4(v2x, v2y, v2z, v2w, u1x, u1y, u1z, u1w);
      const float g2 = DOT4(v2x, v2y, v2z, v2w, u2x, u2y, u2z, u2w);
      const float g3 = DOT4(v2x, v2y, v2z, v2w, u3x, u3y, u3z, u3w);
      H[2] = fmaf(g0 * g0, w0, fmaf(g1 * g1, r0, fmaf(g2 * g2, r1, g3 * g3 * r2)));
    }
#undef DOT4

    // Z2 = reverse(z2), Z1 = reverse(z1)
    const float z2dotH = fmaf(s_z2[s * 3 + 2], H[0],
                         fmaf(s_z2[s * 3 + 1], H[1], s_z2[s * 3 + 0] * H[2]));
    const float z1dot  = fmaf(s_z1[s * 3 + 2], dF0,
                         fmaf(s_z1[s * 3 + 1], dF1_, s_z1[s * 3 + 0] * dF2));

    const float cross = __builtin_logf(s_F2[s]) - z2dotH;
    const float ent1  = __builtin_logf(F1v) - z1dot * invF1;
    out[base + s] = cross - ent1;
  }
}

extern "C" void kernel_launch(void* const* d_in, const int* in_sizes, int n_in,
                              void* d_out, int out_size, void* d_ws, size_t ws_size,
                              hipStream_t stream) {
  const float* q1  = (const float*)d_in[0];
  const float* z1  = (const float*)d_in[1];
  const float* q2  = (const float*)d_in[2];
  const float* z2  = (const float*)d_in[3];
  const float* F1  = (const float*)d_in[4];
  const float* dF1 = (const float*)d_in[5];
  const float* F2  = (const float*)d_in[6];
  float* out = (float*)d_out;
  const int B = out_size;
  const int grid = (B + TILE - 1) / TILE;
  bingham_kl_kernel<<<grid, NTHREADS, 0, stream>>>(q1, z1, q2, z2, F1, dF1, F2, out, B);
}